// RecurrentAttention_56014963474434
// MI455X (gfx1250) — compile-verified
//
#include <hip/hip_runtime.h>

// Problem constants (from reference): B=128, S=8192, D=64
#define B_N 128
#define S_N 8192
#define D_N 64
#define NCHUNK 8                      // S-chunks per batch -> 1024 workgroups in pass 1
#define CS (S_N / NCHUNK)             // 1024 rows per chunk
#define TILES_PER_CHUNK (CS / 16)     // 64 16-row WMMA tiles per chunk
#define NWAVES 8                      // 256 threads = 8 wave32
#define PART_STRIDE (2 + D_N)         // per-(b,chunk) partial: m, l, acc[64]

typedef __attribute__((ext_vector_type(2))) float v2f;
typedef __attribute__((ext_vector_type(4))) float v4f;
typedef __attribute__((ext_vector_type(8))) float v8f;

// d_ws layout (floats): partials then per-batch stats
#define WS_PART  0
#define WS_STATS (B_N * NCHUNK * PART_STRIDE)   // 67584

// ---------------------------------------------------------------------------
// Pass 1: one workgroup per (chunk, batch). Streams its 1024 x-rows ONCE.
//   - scores via V_WMMA_F32_16X16X4_F32 (A = x tile, B = v replicated over N)
//   - flash-style online softmax stats (m, l) + weighted accumulator acc[64]
//   - raw scores written into the alpha region of d_out (converted in pass 3)
// ---------------------------------------------------------------------------
__global__ __launch_bounds__(256)
void ra_pass1(const float* __restrict__ x, const float* __restrict__ v,
              float* __restrict__ scores_out, float* __restrict__ ws) {
  const int chunk = blockIdx.x;
  const int b     = blockIdx.y;
  const int tid   = threadIdx.x;
  const int lane  = tid & 31;
  const int wave  = tid >> 5;
  const int half  = lane >> 4;   // 0: rows 0-7 / K 0-1 side, 1: rows 8-15 / K 2-3 side
  const int lrow  = lane & 15;   // M index for the A-matrix

  __shared__ float sv[D_N];
  __shared__ float red_m[NWAVES];
  __shared__ float red_l[NWAVES];
  __shared__ float red_acc[NWAVES][D_N];

  if (tid < D_N) sv[tid] = v[tid];
  __syncthreads();

  const float* xb     = x + (size_t)b * S_N * D_N;
  float*       scores = scores_out + (size_t)b * S_N;

  float m_w  = -3.402823466e38f;   // running max (wave-uniform after shuffles)
  float l_w  = 0.f;                // running sum of exp
  float acc0 = 0.f, acc1 = 0.f;    // lane owns out dims d = 2*lane, 2*lane+1

  for (int t = wave; t < TILES_PER_CHUNK; t += NWAVES) {
    const int s0 = chunk * CS + t * 16;

    // prefetch next tile for this wave (speculative, dropped if OOB)
    if (t + NWAVES < TILES_PER_CHUNK)
      __builtin_prefetch(xb + (size_t)(s0 + NWAVES * 16 + lrow) * D_N, 0, 1);

    // ---- scores: 16 chained fp32 WMMAs reduce K = 64 --------------------
    // A 16x4 f32 layout: lanes 0-15 -> {K0,K1}, lanes 16-31 -> {K2,K3}
    const float* xrow = xb + (size_t)(s0 + lrow) * D_N + 2 * half;
    v8f c = {0.f, 0.f, 0.f, 0.f, 0.f, 0.f, 0.f, 0.f};
#pragma unroll
    for (int k = 0; k < D_N; k += 4) {
      v2f a = *(const v2f*)(xrow + k);
      v2f bb;
      bb.x = sv[k + 2 * half];        // B rows K replicated across all N cols
      bb.y = sv[k + 2 * half + 1];
      c = __builtin_amdgcn_wmma_f32_16x16x4_f32(
          /*neg_a=*/false, a, /*neg_b=*/false, bb,
          /*c_mod=*/(short)0, c, /*reuse_a=*/false, /*reuse_b=*/false);
    }
    // c[r] == score(s0 + 8*half + r), replicated across the 16 lanes of a half

    // ---- store raw scores (lanes 0 and 16 carry both halves) ------------
    if (lrow == 0) {
      v4f lo = {c[0], c[1], c[2], c[3]};
      v4f hi = {c[4], c[5], c[6], c[7]};
      float* sp = scores + s0 + 8 * half;
      *(v4f*)(sp)     = lo;
      *(v4f*)(sp + 4) = hi;
    }

    // ---- online softmax update ------------------------------------------
    float tm = c[0];
#pragma unroll
    for (int r = 1; r < 8; ++r) tm = fmaxf(tm, c[r]);
    tm = fmaxf(tm, __shfl_xor(tm, 16, 32));          // combine the two halves
    const float m_new = fmaxf(m_w, tm);
    const float scale = __expf(m_w - m_new);

    float w[8];
    float lsum = 0.f;
#pragma unroll
    for (int r = 0; r < 8; ++r) { w[r] = __expf(c[r] - m_new); lsum += w[r]; }
    lsum += __shfl_xor(lsum, 16, 32);
    l_w  = l_w * scale + lsum;
    acc0 *= scale;
    acc1 *= scale;
    m_w = m_new;

    // ---- weighted accumulation: coalesced re-read (WGP$ hit) ------------
    const float* xt = xb + (size_t)s0 * D_N + 2 * lane;
#pragma unroll
    for (int r = 0; r < 16; ++r) {
      v2f xv = *(const v2f*)(xt + (size_t)r * D_N);
      const float wr = __shfl(w[r & 7], (r >> 3) << 4, 32);  // rows 0-7 from lane 0, 8-15 from lane 16
      acc0 = fmaf(wr, xv.x, acc0);
      acc1 = fmaf(wr, xv.y, acc1);
    }
  }

  // ---- combine the 8 waves of this workgroup ----------------------------
  if (lane == 0) { red_m[wave] = m_w; red_l[wave] = l_w; }
  red_acc[wave][2 * lane]     = acc0;
  red_acc[wave][2 * lane + 1] = acc1;
  __syncthreads();

  if (tid < D_N) {
    float m = red_m[0];
#pragma unroll
    for (int wv = 1; wv < NWAVES; ++wv) m = fmaxf(m, red_m[wv]);
    float l = 0.f, a = 0.f;
#pragma unroll
    for (int wv = 0; wv < NWAVES; ++wv) {
      const float e = __expf(red_m[wv] - m);
      l += red_l[wv] * e;
      a += red_acc[wv][tid] * e;
    }
    float* p = ws + WS_PART + ((size_t)b * NCHUNK + chunk) * PART_STRIDE;
    p[2 + tid] = a;
    if (tid == 0) { p[0] = m; p[1] = l; }
  }
}

// ---------------------------------------------------------------------------
// Pass 2: one workgroup (64 threads) per batch; combine 8 chunk partials,
// emit out[b, :] and the final (m, l) stats for the alpha pass.
// ---------------------------------------------------------------------------
__global__ __launch_bounds__(64)
void ra_pass2(float* __restrict__ ws, float* __restrict__ out) {
  const int b = blockIdx.x;
  const int d = threadIdx.x;
  const float* part = ws + WS_PART + (size_t)b * NCHUNK * PART_STRIDE;

  float m = -3.402823466e38f;
#pragma unroll
  for (int c = 0; c < NCHUNK; ++c) m = fmaxf(m, part[c * PART_STRIDE]);
  float l = 0.f, acc = 0.f;
#pragma unroll
  for (int c = 0; c < NCHUNK; ++c) {
    const float e = __expf(part[c * PART_STRIDE] - m);
    l   += part[c * PART_STRIDE + 1] * e;
    acc += part[c * PART_STRIDE + 2 + d] * e;
  }
  out[(size_t)b * D_N + d] = acc / l;
  if (d == 0) {
    ws[WS_STATS + 2 * b]     = m;
    ws[WS_STATS + 2 * b + 1] = l;
  }
}

// ---------------------------------------------------------------------------
// Pass 3: elementwise, in place: alpha[i] = exp(score[i] - m_b) / l_b
// ---------------------------------------------------------------------------
__global__ __launch_bounds__(256)
void ra_pass3(const float* __restrict__ ws, float* __restrict__ alpha) {
  const size_t i = (size_t)blockIdx.x * 256 + threadIdx.x;  // i < B*S
  const int b = (int)(i >> 13);                             // S = 8192
  const float m = ws[WS_STATS + 2 * b];
  const float l = ws[WS_STATS + 2 * b + 1];
  alpha[i] = __expf(alpha[i] - m) / l;
}

extern "C" void kernel_launch(void* const* d_in, const int* in_sizes, int n_in,
                              void* d_out, int out_size, void* d_ws, size_t ws_size,
                              hipStream_t stream) {
  const float* x = (const float*)d_in[0];   // (B, S, D) fp32
  const float* v = (const float*)d_in[1];   // (D, 1)   fp32
  float* out   = (float*)d_out;             // [0 : B*D) = out
  float* alpha = out + (size_t)B_N * D_N;   // [B*D : B*D + B*S) = alpha (scores scratch first)
  float* ws    = (float*)d_ws;              // partials + stats (~271 KB)

  dim3 g1(NCHUNK, B_N);
  ra_pass1<<<g1, 256, 0, stream>>>(x, v, alpha, ws);
  ra_pass2<<<B_N, 64, 0, stream>>>(ws, out);
  ra_pass3<<<(B_N * S_N) / 256, 256, 0, stream>>>(ws, alpha);
}